// RVSoftmax_36421322670459
// MI455X (gfx1250) — compile-verified
//
#include <hip/hip_runtime.h>

typedef __attribute__((ext_vector_type(2))) float v2f;
typedef __attribute__((ext_vector_type(8))) float v8f;

#define B_N 256
#define F_N 512
#define TSTRIDE 516   // 516 % 64 banks == 4 -> conflict-free 16-lane column gathers

// ---------------------------------------------------------------------------
// K1: s = softmax(mu, axis=-1)   (one block per sample row)
// ---------------------------------------------------------------------------
__global__ __launch_bounds__(256) void softmax_rows_kernel(const float* __restrict__ mu,
                                                           float* __restrict__ s_ws) {
    __shared__ float red[256];
    const int b = blockIdx.x;
    const int t = threadIdx.x;
    const float x0 = mu[b * F_N + t];
    const float x1 = mu[b * F_N + t + 256];
    float m = fmaxf(x0, x1);
    red[t] = m;
    __syncthreads();
    for (int off = 128; off > 0; off >>= 1) {
        if (t < off) red[t] = fmaxf(red[t], red[t + off]);
        __syncthreads();
    }
    m = red[0];
    __syncthreads();
    const float e0 = __expf(x0 - m), e1 = __expf(x1 - m);
    red[t] = e0 + e1;
    __syncthreads();
    for (int off = 128; off > 0; off >>= 1) {
        if (t < off) red[t] += red[t + off];
        __syncthreads();
    }
    const float inv = 1.0f / red[0];
    s_ws[b * F_N + t]       = e0 * inv;
    s_ws[b * F_N + t + 256] = e1 * inv;
}

// ---------------------------------------------------------------------------
// K2: mu_out = softmax(mu, axis=0)  (one thread per column, online softmax;
//     consecutive threads read consecutive addresses -> coalesced)
// ---------------------------------------------------------------------------
__global__ __launch_bounds__(256) void softmax_cols_kernel(const float* __restrict__ mu,
                                                           float* __restrict__ mu_out) {
    const int c = blockIdx.x * 256 + threadIdx.x;   // 0..511
    float m = -3.402823466e38f;
    float sum = 0.f;
    for (int r = 0; r < B_N; ++r) {
        const float x = mu[r * F_N + c];
        const float nm = fmaxf(m, x);
        sum = sum * __expf(m - nm) + __expf(x - nm);
        m = nm;
    }
    const float inv = 1.0f / sum;
    for (int r = 0; r < B_N; ++r)
        mu_out[r * F_N + c] = __expf(mu[r * F_N + c] - m) * inv;
}

// ---------------------------------------------------------------------------
// K3: per sample b: u = Sigma_b s_b, w = Sigma_b^T s_b, alpha = s_b . u
//     Streams Sigma_b once through LDS; both reductions done with
//     V_WMMA_F32_16X16X4_F32 (vector replicated across the free dimension).
// ---------------------------------------------------------------------------
__global__ __launch_bounds__(256) void uw_alpha_kernel(const float* __restrict__ Sigma,
                                                       const float* __restrict__ s_ws,
                                                       float* __restrict__ u_ws,
                                                       float* __restrict__ w_ws,
                                                       float* __restrict__ alpha_ws) {
    __shared__ float s_sm[F_N];
    __shared__ float u_sm[F_N];
    __shared__ float tile[16 * TSTRIDE];
    __shared__ float red[8];

    const int b    = blockIdx.x;
    const int t    = threadIdx.x;
    const int wv   = t >> 5;        // wave id 0..7 (wave32)
    const int lane = t & 31;
    const int lm   = lane & 15;     // M for A-tiles, N for B/D tiles
    const int lh   = lane >> 4;     // K-half selector
    const int col0 = wv * 64;       // this wave owns columns [col0, col0+64)

    s_sm[t]       = s_ws[b * F_N + t];
    s_sm[t + 256] = s_ws[b * F_N + t + 256];
    u_sm[t] = 0.f;
    u_sm[t + 256] = 0.f;
    __syncthreads();

    v8f Dw[4] = {};                 // persistent w-accumulators (4 x 16 columns)
    const float* Sb = Sigma + (size_t)b * F_N * F_N;

    for (int c = 0; c < 32; ++c) {
        const int i0 = c * 16;      // row chunk [i0, i0+16)

        // stage 16x512 chunk into LDS (coalesced float4 loads)
        for (int idx = t; idx < 2048; idx += 256) {
            const int row  = idx >> 7;
            const int col4 = (idx & 127) << 2;
            const float4 v = *(const float4*)(Sb + (size_t)(i0 + row) * F_N + col4);
            *(float4*)(&tile[row * TSTRIDE + col4]) = v;
        }
        __syncthreads();

        // ---- u path: D = Sigma_tile(16x4) x s_repl(4x16), accumulate over K ----
        v8f Du = {};
        for (int k0 = 0; k0 < 64; k0 += 4) {
            const int cidx = col0 + k0 + 2 * lh;
            v2f A  = { tile[lm * TSTRIDE + cidx], tile[lm * TSTRIDE + cidx + 1] };
            v2f Bv = { s_sm[cidx], s_sm[cidx + 1] };
            Du = __builtin_amdgcn_wmma_f32_16x16x4_f32(false, A, false, Bv,
                                                       (short)0, Du, false, false);
        }
        // column 0 of D lives in lanes with N==0 (lanes 0 and 16)
        if (lm == 0) {
            #pragma unroll
            for (int v = 0; v < 8; ++v)
                atomicAdd(&u_sm[i0 + v + 8 * lh], Du[v]);
        }

        // ---- w path: D = s_repl(16x4) x Sigma_tile(4x16), accumulate over rows ----
        #pragma unroll
        for (int cg = 0; cg < 4; ++cg) {
            const int cb = col0 + cg * 16;
            #pragma unroll
            for (int k = 0; k < 16; k += 4) {
                const int r0 = k + 2 * lh;
                v2f A  = { s_sm[i0 + r0], s_sm[i0 + r0 + 1] };
                v2f Bv = { tile[r0 * TSTRIDE + cb + lm],
                           tile[(r0 + 1) * TSTRIDE + cb + lm] };
                Dw[cg] = __builtin_amdgcn_wmma_f32_16x16x4_f32(false, A, false, Bv,
                                                               (short)0, Dw[cg], false, false);
            }
        }
        __syncthreads();
    }

    // write w: every lane of the low half holds one column (M=0 row of D)
    if (lh == 0) {
        w_ws[b * F_N + col0 +  0 + lm] = Dw[0][0];
        w_ws[b * F_N + col0 + 16 + lm] = Dw[1][0];
        w_ws[b * F_N + col0 + 32 + lm] = Dw[2][0];
        w_ws[b * F_N + col0 + 48 + lm] = Dw[3][0];
    }

    // write u, reduce alpha = s . u
    const float u0 = u_sm[t], u1 = u_sm[t + 256];
    u_ws[b * F_N + t]       = u0;
    u_ws[b * F_N + t + 256] = u1;
    float part = s_sm[t] * u0 + s_sm[t + 256] * u1;
    for (int off = 16; off > 0; off >>= 1)
        part += __shfl_down(part, off, 32);
    if (lane == 0) red[wv] = part;
    __syncthreads();
    if (t == 0) {
        float a = 0.f;
        #pragma unroll
        for (int i = 0; i < 8; ++i) a += red[i];
        alpha_ws[b] = a;
    }
}

// ---------------------------------------------------------------------------
// K4: Sigma_out[b][i][j] = s_i s_j (Sigma_ij - u_i - w_j + alpha)
//     One block per (b,i) row; fully coalesced float4 stream.
// ---------------------------------------------------------------------------
__global__ __launch_bounds__(128) void finalize_kernel(const float* __restrict__ Sigma,
                                                       const float* __restrict__ s_ws,
                                                       const float* __restrict__ u_ws,
                                                       const float* __restrict__ w_ws,
                                                       const float* __restrict__ alpha_ws,
                                                       float* __restrict__ Sigma_out) {
    const int bi = blockIdx.x;       // 0 .. B*F-1
    const int b  = bi >> 9;
    const int i  = bi & (F_N - 1);
    const float si = s_ws[b * F_N + i];
    const float ui = u_ws[b * F_N + i];
    const float c0 = alpha_ws[b] - ui;

    const int j = threadIdx.x << 2;
    const size_t base = (size_t)bi * F_N + j;
    const float4 Sv = *(const float4*)(Sigma + base);
    const float4 sj = *(const float4*)(s_ws + b * F_N + j);
    const float4 wj = *(const float4*)(w_ws + b * F_N + j);
    float4 o;
    o.x = si * sj.x * (Sv.x - wj.x + c0);
    o.y = si * sj.y * (Sv.y - wj.y + c0);
    o.z = si * sj.z * (Sv.z - wj.z + c0);
    o.w = si * sj.w * (Sv.w - wj.w + c0);
    *(float4*)(Sigma_out + base) = o;
}

// ---------------------------------------------------------------------------
extern "C" void kernel_launch(void* const* d_in, const int* in_sizes, int n_in,
                              void* d_out, int out_size, void* d_ws, size_t ws_size,
                              hipStream_t stream) {
    const float* mu    = (const float*)d_in[0];   // (256, 512)
    const float* Sigma = (const float*)d_in[1];   // (256, 512, 512)
    float* out       = (float*)d_out;
    float* mu_out    = out;                       // first B*F floats
    float* Sigma_out = out + B_N * F_N;           // then B*F*F floats

    float* ws       = (float*)d_ws;
    float* s_ws     = ws;                         // B*F
    float* u_ws     = ws + 1 * B_N * F_N;         // B*F
    float* w_ws     = ws + 2 * B_N * F_N;         // B*F
    float* alpha_ws = ws + 3 * B_N * F_N;         // B

    softmax_rows_kernel<<<B_N, 256, 0, stream>>>(mu, s_ws);
    softmax_cols_kernel<<<F_N / 256, 256, 0, stream>>>(mu, mu_out);
    uw_alpha_kernel<<<B_N, 256, 0, stream>>>(Sigma, s_ws, u_ws, w_ws, alpha_ws);
    finalize_kernel<<<B_N * F_N, 128, 0, stream>>>(Sigma, s_ws, u_ws, w_ws, alpha_ws,
                                                   Sigma_out);
}